// RNNModel_88467736363729
// MI455X (gfx1250) — compile-verified
//
#include <hip/hip_runtime.h>
#include <hip/hip_bf16.h>
#include <math.h>

typedef __attribute__((ext_vector_type(2))) float v2f;
typedef __attribute__((ext_vector_type(8))) float v8f;

#define WMMA_F32X4(a, b, c) \
  __builtin_amdgcn_wmma_f32_16x16x4_f32(false, (a), false, (b), (short)0, (c), false, false)

__device__ __forceinline__ float sigm(float v) { return 1.f / (1.f + __expf(-v)); }
__device__ __forceinline__ float actf(float v, int a) {
  return (a == 0) ? fmaxf(v, 0.f) : ((a == 1) ? tanhf(v) : v);
}
__device__ __forceinline__ v2f ld2(const float* p) { return *(const v2f*)p; }

// ---------------------------------------------------------------------------
// 1x1 conv as WMMA GEMM: y[b, co_off+m, lo] = sum_ci act(x[b, ci, lo*stride+lshift]) * w[m, ci]
// A = w (Cout x Cin), B = x gathered as (Cin x B*Lout). One wave per 16x16 tile.
// Branch-free: clamped indices + 0/1 mask multiplies; RELU is a template param
// so the non-relu instantiation has a pure load->mul->wmma loop.
// ---------------------------------------------------------------------------
template <int RELU>
__global__ __launch_bounds__(32) void k_conv1x1(
    const float* __restrict__ x, const float* __restrict__ w, float* __restrict__ y,
    int B, int Cin, int Lin, int Cout, int Lout, int stride, int lshift,
    int co_off, int CoutTot) {
  const int lane = threadIdx.x;
  const int n0 = blockIdx.x * 16, m0 = blockIdx.y * 16;
  const int N = B * Lout;
  const int mA = m0 + (lane & 15);
  const int nB = n0 + (lane & 15);
  const int kHi = (lane >> 4) * 2;
  const float aM = (mA < Cout) ? 1.f : 0.f;
  const int mC = (mA < Cout) ? mA : (Cout - 1);
  const int nC = (nB < N) ? nB : (N - 1);
  const int bIdx = nC / Lout, lo = nC % Lout;
  int li = lo * stride + lshift;
  const float bM = ((nB < N) && (li < Lin)) ? 1.f : 0.f;
  if (li >= Lin) li = Lin - 1;
  const float* wrow = w + (size_t)mC * Cin;      // contiguous, 8B aligned (Cin % 4 == 0)
  const float* xcol = x + ((size_t)bIdx * Cin) * (size_t)Lin + li;
  v8f acc = {};
  for (int k0 = 0; k0 < Cin; k0 += 4) {          // Cin is a multiple of 4 in all uses
    const int ka = k0 + kHi;
    v2f a = ld2(wrow + ka);
    a.x *= aM; a.y *= aM;
    float bx = xcol[(size_t)ka * Lin];
    float by = xcol[(size_t)(ka + 1) * Lin];
    if (RELU) { bx = fmaxf(bx, 0.f); by = fmaxf(by, 0.f); }
    v2f bf; bf.x = bx * bM; bf.y = by * bM;
    acc = WMMA_F32X4(a, bf, acc);
  }
  const int n = n0 + (lane & 15);
  if (n < N) {
    const int bO = n / Lout, lO = n % Lout;
    float* ybase = y + ((size_t)bO * CoutTot + co_off) * (size_t)Lout + lO;
#pragma unroll
    for (int r = 0; r < 8; ++r) {
      const int m = m0 + r + 8 * (lane >> 4);
      if (m < Cout) ybase[(size_t)m * Lout] = acc[r];
    }
  }
}

// ---------------------------------------------------------------------------
// Generic WMMA GEMM, 16x32 tile per wave (2 accumulators share one A fragment):
//   C[m,n] = act(sum_k A[m*lda + k] * Bm[n*ldbn + k] + bias[n])
// Branch-free main loop + single masked tail iteration for K % 4.
// ---------------------------------------------------------------------------
__global__ __launch_bounds__(32) void k_gemm_bias_act(
    const float* __restrict__ A, long lda,
    const float* __restrict__ Bm, long ldbn,
    const float* __restrict__ bias, float* __restrict__ Cq, long ldc,
    int M, int N, int K, int act) {
  const int lane = threadIdx.x;
  const int n0 = blockIdx.x * 32, m0 = blockIdx.y * 16;
  const int mA = m0 + (lane & 15);
  const int nB0 = n0 + (lane & 15);
  const int nB1 = nB0 + 16;
  const int kHi = (lane >> 4) * 2;
  const float aM = (mA < M) ? 1.f : 0.f;
  const float bM0 = (nB0 < N) ? 1.f : 0.f;
  const float bM1 = (nB1 < N) ? 1.f : 0.f;
  const int mC = (mA < M) ? mA : (M - 1);
  const int nC0 = (nB0 < N) ? nB0 : (N - 1);
  const int nC1 = (nB1 < N) ? nB1 : (N - 1);
  const float* arow = A + (size_t)mC * lda;
  const float* brow0 = Bm + (size_t)nC0 * ldbn;
  const float* brow1 = Bm + (size_t)nC1 * ldbn;
  v8f acc0 = {}, acc1 = {};
  const int Kmain = K & ~3;
  for (int k0 = 0; k0 < Kmain; k0 += 4) {
    const int ka = k0 + kHi;
    v2f a, b0, b1;
    a.x = arow[ka] * aM;       a.y = arow[ka + 1] * aM;
    b0.x = brow0[ka] * bM0;    b0.y = brow0[ka + 1] * bM0;
    b1.x = brow1[ka] * bM1;    b1.y = brow1[ka + 1] * bM1;
    acc0 = WMMA_F32X4(a, b0, acc0);
    acc1 = WMMA_F32X4(a, b1, acc1);
  }
  if (Kmain < K) {  // masked tail (K % 4 != 0)
    const int ka = Kmain + kHi;
    const int k0c = (ka < K) ? ka : (K - 1);
    const int k1c = (ka + 1 < K) ? (ka + 1) : (K - 1);
    const float kM0 = (ka < K) ? 1.f : 0.f;
    const float kM1 = (ka + 1 < K) ? 1.f : 0.f;
    v2f a, b0, b1;
    a.x = arow[k0c] * (aM * kM0);     a.y = arow[k1c] * (aM * kM1);
    b0.x = brow0[k0c] * (bM0 * kM0);  b0.y = brow0[k1c] * (bM0 * kM1);
    b1.x = brow1[k0c] * (bM1 * kM0);  b1.y = brow1[k1c] * (bM1 * kM1);
    acc0 = WMMA_F32X4(a, b0, acc0);
    acc1 = WMMA_F32X4(a, b1, acc1);
  }
  const int nA = n0 + (lane & 15);
#pragma unroll
  for (int half = 0; half < 2; ++half) {
    const int n = nA + 16 * half;
    if (n >= N) continue;
    const float bv = bias ? bias[n] : 0.f;
    const v8f& acc = half ? acc1 : acc0;
#pragma unroll
    for (int r = 0; r < 8; ++r) {
      const int m = m0 + r + 8 * (lane >> 4);
      if (m < M) {
        float v = acc[r] + bv;
        if (act == 1) v = fmaxf(v, 0.f);
        Cq[(size_t)m * ldc + n] = v;
      }
    }
  }
}

// ---------------------------------------------------------------------------
// RNN step GEMM fused with DARTS gating.
// ch = [A0|A1](64 x (K0+K1)) @ W(K x 1024) ; c = sigmoid(ch[:, :512]);
// h = act(ch[:, 512:]) ; out = sp + c*(h - sp).  M=64 fixed, N=512, W ld = 1024.
// ---------------------------------------------------------------------------
__global__ __launch_bounds__(32) void k_rnn_gate(
    const float* __restrict__ A0, const float* __restrict__ A1,
    int K0, int K1, const float* __restrict__ W,
    const float* __restrict__ sp, float* __restrict__ out, int act) {
  const int lane = threadIdx.x;
  const int n0 = blockIdx.x * 16, m0 = blockIdx.y * 16;
  const int mA = m0 + (lane & 15);
  const int nB = n0 + (lane & 15);
  const int kHi = (lane >> 4) * 2;
  v8f accC = {}, accH = {};
  {
    const float* arow = A0 + (size_t)mA * K0;    // K0 even -> 8B aligned
    for (int k0 = 0; k0 < K0; k0 += 4) {
      const int ka = k0 + kHi;
      v2f a = ld2(arow + ka);
      const float* wr = W + (size_t)ka * 1024;
      v2f bc, bh;
      bc.x = wr[nB];        bc.y = wr[1024 + nB];
      bh.x = wr[512 + nB];  bh.y = wr[1024 + 512 + nB];
      accC = WMMA_F32X4(a, bc, accC);
      accH = WMMA_F32X4(a, bh, accH);
    }
  }
  if (K1 > 0) {
    const float* arow = A1 + (size_t)mA * K1;
    const float* Wseg = W + (size_t)K0 * 1024;
    for (int k0 = 0; k0 < K1; k0 += 4) {
      const int ka = k0 + kHi;
      v2f a = ld2(arow + ka);
      const float* wr = Wseg + (size_t)ka * 1024;
      v2f bc, bh;
      bc.x = wr[nB];        bc.y = wr[1024 + nB];
      bh.x = wr[512 + nB];  bh.y = wr[1024 + 512 + nB];
      accC = WMMA_F32X4(a, bc, accC);
      accH = WMMA_F32X4(a, bh, accH);
    }
  }
  const int n = n0 + (lane & 15);
#pragma unroll
  for (int r = 0; r < 8; ++r) {
    const int m = m0 + r + 8 * (lane >> 4);
    const float spv = sp[(size_t)m * 512 + n];
    const float c = sigm(accC[r]);
    const float h = actf(accH[r], act);
    out[(size_t)m * 512 + n] = spv + c * (h - spv);
  }
}

// ---------------------------------------------------------------------------
// Elementwise / reduction helpers.  3D grid: (L-blocks, C, B) -> no div/mod.
// ---------------------------------------------------------------------------
__global__ void k_stem(const float* __restrict__ x, const float* __restrict__ w,
                       float* __restrict__ y, int Cin, int L, int KS, int pad) {
  const int l = blockIdx.x * blockDim.x + threadIdx.x;
  if (l >= L) return;
  const int co = blockIdx.y, b = blockIdx.z;
  const int Cout = gridDim.y;
  float acc = 0.f;
  for (int ci = 0; ci < Cin; ++ci)
    for (int t = 0; t < KS; ++t) {
      const int li = l + t - pad;
      if (li >= 0 && li < L)
        acc += x[((size_t)b * Cin + ci) * L + li] * w[((size_t)co * Cin + ci) * KS + t];
    }
  y[((size_t)b * Cout + co) * L + l] = acc;
}

__global__ void k_bn_stats(const float* __restrict__ x, float* __restrict__ stats,
                           int B, int C, int L, int lgL) {
  __shared__ float sh[256], sh2[256];
  const int c = blockIdx.x;
  const int n = B << lgL;          // B * L, L = 1<<lgL
  const int lmask = L - 1;
  float s = 0.f, ss = 0.f;
  for (int i = threadIdx.x; i < n; i += blockDim.x) {
    const int b = i >> lgL, l = i & lmask;
    const float v = x[((size_t)b * C + c) * L + l];
    s += v; ss += v * v;
  }
  sh[threadIdx.x] = s; sh2[threadIdx.x] = ss;
  __syncthreads();
  for (int o = 128; o > 0; o >>= 1) {
    if ((int)threadIdx.x < o) { sh[threadIdx.x] += sh[threadIdx.x + o]; sh2[threadIdx.x] += sh2[threadIdx.x + o]; }
    __syncthreads();
  }
  if (threadIdx.x == 0) {
    const float m = sh[0] / (float)n;
    const float var = sh2[0] / (float)n - m * m;
    stats[c] = m;
    stats[C + c] = rsqrtf(var + 1e-5f);
  }
}

__global__ void k_bn_apply(float* __restrict__ x, const float* __restrict__ stats,
                           const float* __restrict__ g, const float* __restrict__ bb,
                           int L) {
  const int l = blockIdx.x * blockDim.x + threadIdx.x;
  if (l >= L) return;
  const int c = blockIdx.y, b = blockIdx.z;
  const int C = gridDim.y;
  const size_t i = ((size_t)b * C + c) * L + l;
  float y = (x[i] - stats[c]) * stats[C + c];
  if (g) y = y * g[c] + bb[c];
  x[i] = y;
}

__global__ void k_dwconv3(const float* __restrict__ x, int CtIn,
                          const float* __restrict__ w, float* __restrict__ y,
                          int Lin, int Lout, int stride, int relu_in) {
  const int lo = blockIdx.x * blockDim.x + threadIdx.x;
  if (lo >= Lout) return;
  const int c = blockIdx.y, b = blockIdx.z;
  const int C = gridDim.y;
  const float* xr = x + ((size_t)b * CtIn + c) * (size_t)Lin;
  float acc = 0.f;
#pragma unroll
  for (int t = 0; t < 3; ++t) {
    const int li = lo * stride + t - 1;
    if (li >= 0 && li < Lin) {
      float v = xr[li];
      if (relu_in) v = fmaxf(v, 0.f);
      acc += v * w[c * 3 + t];
    }
  }
  y[((size_t)b * C + c) * (size_t)Lout + lo] = acc;
}

__global__ void k_maxpool3(const float* __restrict__ x, int CtIn, float* __restrict__ y,
                           int Lin, int Lout, int stride) {
  const int lo = blockIdx.x * blockDim.x + threadIdx.x;
  if (lo >= Lout) return;
  const int c = blockIdx.y, b = blockIdx.z;
  const int C = gridDim.y;
  const float* xr = x + ((size_t)b * CtIn + c) * (size_t)Lin;
  float m = -INFINITY;
#pragma unroll
  for (int t = 0; t < 3; ++t) {
    const int li = lo * stride + t - 1;
    if (li >= 0 && li < Lin) m = fmaxf(m, xr[li]);
  }
  y[((size_t)b * C + c) * (size_t)Lout + lo] = m;
}

__global__ void k_edge_add(float* __restrict__ acc, int CtAcc,
                           const float* __restrict__ xskip, int CtX,
                           const float* __restrict__ sep, int L, int init) {
  const int l = blockIdx.x * blockDim.x + threadIdx.x;
  if (l >= L) return;
  const int c = blockIdx.y, b = blockIdx.z;
  const int C = gridDim.y;
  const float v = xskip[((size_t)b * CtX + c) * (size_t)L + l] +
                  sep[((size_t)b * C + c) * (size_t)L + l];
  float* a = acc + ((size_t)b * CtAcc + c) * (size_t)L + l;
  *a = init ? v : (*a + v);
}

__global__ void k_mean8(const float* s1, const float* s2, const float* s3, const float* s4,
                        const float* s5, const float* s6, const float* s7, const float* s8,
                        float* __restrict__ out) {
  const int i = blockIdx.x * blockDim.x + threadIdx.x;
  if (i >= 64 * 512) return;
  out[i] = (s1[i] + s2[i] + s3[i] + s4[i] + s5[i] + s6[i] + s7[i] + s8[i]) * 0.125f;
}

// X[t,b,h] = s1[b,h,t]   grid: (512/256, B=64, T=128)
__global__ void k_make_X(const float* __restrict__ s1, float* __restrict__ X) {
  const int h = blockIdx.x * blockDim.x + threadIdx.x;
  const int b = blockIdx.y, t = blockIdx.z;
  X[((size_t)t * 64 + b) * 512 + h] = s1[((size_t)b * 512 + h) * 128 + t];
}

// flat[b, t*512+h] = hiddens[t,b,h]   grid: (512/256, T=128, B=64)
__global__ void k_make_flat(const float* __restrict__ hid, float* __restrict__ flat) {
  const int h = blockIdx.x * blockDim.x + threadIdx.x;
  const int t = blockIdx.y, b = blockIdx.z;
  flat[(size_t)b * 65536 + t * 512 + h] = hid[((size_t)t * 64 + b) * 512 + h];
}

// ---------------------------------------------------------------------------
// Host orchestration
// ---------------------------------------------------------------------------
struct EdgeP { const float *dw1, *dw2, *pw1, *pw2; };
struct CellP { const EdgeP* e; const float* pre0_w; const float* pre0_w1;
               const float* pre0_w2; const float* pre1_w; };

static inline int ilog2(int v) { int r = 0; while ((1 << r) < v) ++r; return r; }
static inline dim3 bcl(int L, int C, int B) { return dim3((L + 255) / 256, C, B); }

static void run_bn(float* x, int B, int C, int L, const float* g, const float* bb,
                   float* stats, hipStream_t st) {
  k_bn_stats<<<C, 256, 0, st>>>(x, stats, B, C, L, ilog2(L));
  k_bn_apply<<<bcl(L, C, B), 256, 0, st>>>(x, stats, g, bb, L);
}

static void run_conv1x1(const float* x, const float* w, float* y, int B, int Cin, int Lin,
                        int Cout, int Lout, int stride, int lshift, int co_off, int CoutTot,
                        int relu_in, hipStream_t st) {
  dim3 grid((B * Lout + 15) / 16, (Cout + 15) / 16);
  if (relu_in)
    k_conv1x1<1><<<grid, 32, 0, st>>>(x, w, y, B, Cin, Lin, Cout, Lout, stride, lshift,
                                      co_off, CoutTot);
  else
    k_conv1x1<0><<<grid, 32, 0, st>>>(x, w, y, B, Cin, Lin, Cout, Lout, stride, lshift,
                                      co_off, CoutTot);
}

static void run_sep3(const float* x, int CtX, int Lx, const EdgeP& ep, int B, int C,
                     int stride, int Lout, float* tmpA, float* tmpB, float* outSep,
                     float* stats, hipStream_t st) {
  k_dwconv3<<<bcl(Lout, C, B), 256, 0, st>>>(x, CtX, ep.dw1, tmpA, Lx, Lout, stride, 1);
  run_conv1x1(tmpA, ep.pw1, tmpB, B, C, Lout, C, Lout, 1, 0, 0, C, 0, st);
  run_bn(tmpB, B, C, Lout, nullptr, nullptr, stats, st);
  k_dwconv3<<<bcl(Lout, C, B), 256, 0, st>>>(tmpB, C, ep.dw2, tmpA, Lout, Lout, 1, 1);
  run_conv1x1(tmpA, ep.pw2, outSep, B, C, Lout, C, Lout, 1, 0, 0, C, 0, st);
  run_bn(outSep, B, C, Lout, nullptr, nullptr, stats, st);
}

static void run_cell(const float* s0in, int C0in, int L0,
                     const float* s1in, int C1in, int L1,
                     const CellP& cp, int C, bool red, bool red_prev,
                     float* out,  // (B, 4C, Lout) slab; states 2..5 live here
                     float* s0p, float* s1p, float* tmpA, float* tmpB,
                     float* tmpSep, float* tmpPool, float* stats, hipStream_t st) {
  const int B = 64;
  const int Lp = L1;
  if (red_prev) {  // fact_reduce: relu -> two stride-2 1x1 convs -> bn
    run_conv1x1(s0in, cp.pre0_w1, s0p, B, C0in, L0, C / 2, Lp, 2, 0, 0,     C, 1, st);
    run_conv1x1(s0in, cp.pre0_w2, s0p, B, C0in, L0, C / 2, Lp, 2, 1, C / 2, C, 1, st);
    run_bn(s0p, B, C, Lp, nullptr, nullptr, stats, st);
  } else {
    run_conv1x1(s0in, cp.pre0_w, s0p, B, C0in, L0, C, Lp, 1, 0, 0, C, 1, st);
    run_bn(s0p, B, C, Lp, nullptr, nullptr, stats, st);
  }
  run_conv1x1(s1in, cp.pre1_w, s1p, B, C1in, L1, C, Lp, 1, 0, 0, C, 1, st);
  run_bn(s1p, B, C, Lp, nullptr, nullptr, stats, st);

  const int Lout = red ? Lp / 2 : Lp;
  const float* statePtr[6]; int stateCt[6]; int stateL[6];
  statePtr[0] = s0p; stateCt[0] = C; stateL[0] = Lp;
  statePtr[1] = s1p; stateCt[1] = C; stateL[1] = Lp;
  int off = 0;
  for (int i = 0; i < 4; ++i) {
    float* acc = out + (size_t)i * C * Lout;  // channel slab i of concat output
    for (int j = 0; j < 2 + i; ++j) {
      const float* xs = statePtr[j];
      const int Ct = stateCt[j], Lx = stateL[j];
      const int strideE = (red && j < 2) ? 2 : 1;
      run_sep3(xs, Ct, Lx, cp.e[off + j], 64, C, strideE, Lout, tmpA, tmpB, tmpSep, stats, st);
      const float* skip; int CtSkip;
      if (red && j < 2) {
        k_maxpool3<<<bcl(Lout, C, B), 256, 0, st>>>(xs, Ct, tmpPool, Lx, Lout, 2);
        run_bn(tmpPool, B, C, Lout, nullptr, nullptr, stats, st);
        skip = tmpPool; CtSkip = C;
      } else { skip = xs; CtSkip = Ct; }
      k_edge_add<<<bcl(Lout, C, B), 256, 0, st>>>(acc, 4 * C, skip, CtSkip, tmpSep,
                                                  Lout, j == 0 ? 1 : 0);
    }
    off += 2 + i;
    statePtr[2 + i] = acc; stateCt[2 + i] = 4 * C; stateL[2 + i] = Lout;
  }
}

extern "C" void kernel_launch(void* const* d_in, const int* in_sizes, int n_in,
                              void* d_out, int out_size, void* d_ws, size_t ws_size,
                              hipStream_t stream) {
  (void)in_sizes; (void)n_in; (void)out_size; (void)ws_size;
  // --- unpack params (jax tree order: top-level dict insertion order, nested sorted keys)
  int p = 0;
  const float* in_x = (const float*)d_in[p++];   // inputs (64,4,512)
  const float* in_h = (const float*)d_in[p++];   // hidden (1,64,512)
  const float* W0 = (const float*)d_in[p++];     // (1024,1024)
  const float* Ws[8];
  for (int i = 0; i < 8; ++i) Ws[i] = (const float*)d_in[p++];  // (512,1024) each
  EdgeP cellsE[3][14];
  const float* pre0w[3] = {nullptr, nullptr, nullptr};
  const float* pre0w1 = nullptr; const float* pre0w2 = nullptr;
  const float* pre1w[3];
  for (int c = 0; c < 3; ++c) {
    for (int j = 0; j < 14; ++j) {  // edge dict keys sorted: dw1, dw2, pw1, pw2
      cellsE[c][j].dw1 = (const float*)d_in[p++];
      cellsE[c][j].dw2 = (const float*)d_in[p++];
      cellsE[c][j].pw1 = (const float*)d_in[p++];
      cellsE[c][j].pw2 = (const float*)d_in[p++];
    }
    if (c == 2) { pre0w1 = (const float*)d_in[p++]; pre0w2 = (const float*)d_in[p++]; }
    else        { pre0w[c] = (const float*)d_in[p++]; }
    pre1w[c] = (const float*)d_in[p++];
  }
  const float* cls_b = (const float*)d_in[p++];
  const float* cls_w = (const float*)d_in[p++];   // (4,925)
  const float* dec_b = (const float*)d_in[p++];
  const float* dec_w = (const float*)d_in[p++];   // (925,65536)
  const float* stem_b = (const float*)d_in[p++];  // stem keys sorted: b, g, w
  const float* stem_g = (const float*)d_in[p++];
  const float* stem_w = (const float*)d_in[p++];  // (96,4,13)

  // --- workspace layout (floats)
  float* w = (float*)d_ws;
  size_t o = 0;
  auto alloc = [&](size_t n) { float* q = w + o; o += n; return q; };
  float* stemBuf = alloc(64L * 96 * 512);
  float* cout0 = alloc(64L * 128 * 512);
  float* cout1 = alloc(64L * 256 * 256);
  float* cout2 = alloc(64L * 512 * 128);
  float* s0p = alloc(64L * 64 * 512);     // max (B,C,Lp) = 2M floats
  float* s1p = alloc(64L * 64 * 512);
  float* tmpA = alloc(64L * 64 * 512);
  float* tmpB = alloc(64L * 64 * 512);
  float* tmpSep = alloc(64L * 64 * 512);
  float* tmpPool = alloc(64L * 64 * 512);
  float* stats = alloc(1024);             // mean + rstd (max 512 channels)
  float* X = alloc(128L * 64 * 512);
  float* hiddens = alloc(128L * 64 * 512);
  float* rs[9];
  for (int i = 0; i < 9; ++i) rs[i] = alloc(64L * 512);
  float* flat = alloc(64L * 65536);
  float* decOut = alloc(64L * 928);       // ldc = 928 keeps cls A-rows 8B aligned

  // --- CNN feature extractor
  k_stem<<<dim3(2, 96, 64), 256, 0, stream>>>(in_x, stem_w, stemBuf, 4, 512, 13, 6);
  run_bn(stemBuf, 64, 96, 512, stem_g, stem_b, stats, stream);

  CellP c0{cellsE[0], pre0w[0], nullptr, nullptr, pre1w[0]};
  run_cell(stemBuf, 96, 512, stemBuf, 96, 512, c0, 32, false, false,
           cout0, s0p, s1p, tmpA, tmpB, tmpSep, tmpPool, stats, stream);
  CellP c1{cellsE[1], pre0w[1], nullptr, nullptr, pre1w[1]};
  run_cell(stemBuf, 96, 512, cout0, 128, 512, c1, 64, true, false,
           cout1, s0p, s1p, tmpA, tmpB, tmpSep, tmpPool, stats, stream);
  CellP c2{cellsE[2], nullptr, pre0w1, pre0w2, pre1w[2]};
  run_cell(cout0, 128, 512, cout1, 256, 256, c2, 128, true, true,
           cout2, s0p, s1p, tmpA, tmpB, tmpSep, tmpPool, stats, stream);

  // --- RNN scan: 128 sequential steps, all GEMMs fused with gating epilogue
  k_make_X<<<dim3(2, 64, 128), 256, 0, stream>>>(cout2, X);
  static const int predTab[8] = {0, 1, 2, 3, 4, 1, 5, 1};
  static const int actTab[8]  = {0, 0, 1, 0, 0, 2, 0, 0};  // relu/tanh/identity
  const dim3 gR(32, 4);
  for (int t = 0; t < 128; ++t) {
    const float* xt = X + (size_t)t * 64 * 512;
    const float* hprev = (t == 0) ? in_h : hiddens + (size_t)(t - 1) * 64 * 512;
    // s0 = h + sigmoid(c)*(tanh(h') - h), ch = [x_t | h] @ W0
    k_rnn_gate<<<gR, 32, 0, stream>>>(xt, hprev, 512, 512, W0, hprev, rs[0], 1);
    for (int i = 0; i < 8; ++i) {
      const float* sp = rs[predTab[i]];
      k_rnn_gate<<<gR, 32, 0, stream>>>(sp, sp, 512, 0, Ws[i], sp, rs[i + 1], actTab[i]);
    }
    k_mean8<<<dim3(128), 256, 0, stream>>>(rs[1], rs[2], rs[3], rs[4], rs[5], rs[6],
                                           rs[7], rs[8], hiddens + (size_t)t * 64 * 512);
  }

  // --- decoder: relu(flat @ dec_w.T + dec_b) @ cls_w.T + cls_b
  k_make_flat<<<dim3(2, 128, 64), 256, 0, stream>>>(hiddens, flat);
  {
    dim3 gd((925 + 31) / 32, (64 + 15) / 16);
    k_gemm_bias_act<<<gd, 32, 0, stream>>>(flat, 65536L, dec_w, 65536L, dec_b,
                                           decOut, 928L, 64, 925, 65536, 1);
  }
  {
    dim3 gc(1, (64 + 15) / 16);
    k_gemm_bias_act<<<gc, 32, 0, stream>>>(decOut, 928L, cls_w, 925L, cls_b,
                                           (float*)d_out, 4L, 64, 4, 925, 0);
  }
}